// QuLinear_1425929142983
// MI455X (gfx1250) — compile-verified
//
#include <hip/hip_runtime.h>
#include <hip/hip_bf16.h>
#include <math.h>

typedef float v2f __attribute__((ext_vector_type(2)));
typedef float v8f __attribute__((ext_vector_type(8)));

// ---------- small complex helpers ----------
struct c32 { float x, y; };
__device__ inline c32 cmul(c32 a, c32 b) { return { a.x*b.x - a.y*b.y, a.x*b.y + a.y*b.x }; }
__device__ inline c32 cadd(c32 a, c32 b) { return { a.x + b.x, a.y + b.y }; }
__device__ inline void mm2(const c32 A[2][2], const c32 B[2][2], c32 C[2][2]) {
#pragma unroll
  for (int i = 0; i < 2; ++i)
#pragma unroll
    for (int j = 0; j < 2; ++j)
      C[i][j] = cadd(cmul(A[i][0], B[0][j]), cmul(A[i][1], B[1][j]));
}

// ---------- kernel 1: per-qubit gates -> 32x32 tensor blocks A1,B1,A2,B2 ----------
// gates layout (floats): [A1re|A1im|B1re|B1im|A2re|A2im|B2re|B2im], each 1024 (=32*32)
__global__ __launch_bounds__(256) void build_gates(const float* __restrict__ w,
                                                   float* __restrict__ gates) {
  __shared__ float gr[2][10][2][2], gi[2][10][2][2];
  int t = threadIdx.x;
  if (t < 20) {
    int sg = t / 10, q = t % 10;
    float sign = sg ? -1.0f : 1.0f;
    const float WM = 0.63245553203367588f;   // sqrt(2/5) = he_std * lrmul
    float ax = 0.5f * sign * WM * w[q];
    float ay = 0.5f * sign * WM * w[q + 10];
    float az = 0.5f * sign * WM * w[q + 20];
    float cx = cosf(ax), sx = sinf(ax);
    float cy = cosf(ay), sy = sinf(ay);
    float cz = cosf(az), sz = sinf(az);
    c32 RX[2][2] = { { {cx, 0.f}, {0.f, -sx} }, { {0.f, -sx}, {cx, 0.f} } };
    c32 RY[2][2] = { { {cy, 0.f}, {-sy, 0.f} }, { {sy, 0.f}, {cy, 0.f} } };
    c32 RZ[2][2] = { { {cz, -sz}, {0.f, 0.f} }, { {0.f, 0.f}, {cz, sz} } };
    c32 T[2][2], M[2][2];
    mm2(RY, RX, T);      // RY*RX
    mm2(RZ, T, M);       // RZ*RY*RX  (reference applies RX, then RY, then RZ)
#pragma unroll
    for (int i = 0; i < 2; ++i)
#pragma unroll
      for (int j = 0; j < 2; ++j) { gr[sg][q][i][j] = M[i][j].x; gi[sg][q][i][j] = M[i][j].y; }
  }
  __syncthreads();
  // A = G0..G4 tensor product (qubit 0 = MSB), B = G5..G9
  for (int e = t; e < 1024; e += 256) {
    int i = e >> 5, j = e & 31;
    c32 a  = {1.f, 0.f}, b  = {1.f, 0.f};
    c32 a2 = {1.f, 0.f}, b2 = {1.f, 0.f};
#pragma unroll
    for (int tt = 0; tt < 5; ++tt) {
      int ib = (i >> (4 - tt)) & 1, jb = (j >> (4 - tt)) & 1;
      a  = cmul(a,  { gr[0][tt][ib][jb],     gi[0][tt][ib][jb]     });
      b  = cmul(b,  { gr[0][5 + tt][ib][jb], gi[0][5 + tt][ib][jb] });
      a2 = cmul(a2, { gr[1][tt][ib][jb],     gi[1][tt][ib][jb]     });
      b2 = cmul(b2, { gr[1][5 + tt][ib][jb], gi[1][5 + tt][ib][jb] });
    }
    gates[0 * 1024 + e] = a.x;  gates[1 * 1024 + e] = a.y;
    gates[2 * 1024 + e] = b.x;  gates[3 * 1024 + e] = b.y;
    gates[4 * 1024 + e] = a2.x; gates[5 * 1024 + e] = a2.y;
    gates[6 * 1024 + e] = b2.x; gates[7 * 1024 + e] = b2.y;
  }
}

// ---------- kernel 2: T1 = A1 (x) B1, elementwise (1024x1024 complex) ----------
__global__ __launch_bounds__(256) void build_T1(const float* __restrict__ gates,
                                                float* __restrict__ outre,
                                                float* __restrict__ outim) {
  int idx = blockIdx.x * 256 + threadIdx.x;
  int i = idx >> 10, j = idx & 1023;
  int ea = (i >> 5) * 32 + (j >> 5);
  int eb = (i & 31) * 32 + (j & 31);
  float arx = gates[ea],        aix = gates[1024 + ea];
  float brx = gates[2048 + eb], bix = gates[3072 + eb];
  outre[idx] = arx * brx - aix * bix;
  outim[idx] = arx * bix + aix * brx;
}

// ---------- kernel 3: block-GEMM gate application via V_WMMA_F32_16X16X4_F32 ----------
// MODE_A=0: out[32g+r, c] = sum_k G[r][k] * in[perm(32g+k), c]   (I (x) G), perm=gray if GRAY
// MODE_A=1: out[32r+g, c] = sum_k G[r][k] * in[32k+g, c]         (G (x) I)
// REAL_IN=1: input imaginary part is identically zero (first pass on X).
template <int MODE_A, int GRAY, int REAL_IN>
__global__ __launch_bounds__(128) void apply_gate(float* __restrict__ outre,
                                                  float* __restrict__ outim,
                                                  const float* __restrict__ inre,
                                                  const float* __restrict__ inim,
                                                  const float* __restrict__ Gre,
                                                  const float* __restrict__ Gim) {
  __shared__ float gsr[32][33];   // padded: stride 33 avoids LDS bank conflicts
  __shared__ float gsi[32][33];
  int tid = threadIdx.x;
  for (int t = tid; t < 1024; t += 128) {
    gsr[t >> 5][t & 31] = Gre[t];
    gsi[t >> 5][t & 31] = Gim[t];
  }
  __syncthreads();

  int lane = tid & 31, wv = tid >> 5;
  int g  = blockIdx.x;                 // group (row-block for mode B, residual for mode A)
  int ct = blockIdx.y * 4 + wv;        // 16-column tile index (0..63)
  int l15 = lane & 15, half = lane >> 4;
  int col = ct * 16 + l15;

  v8f zero = {0.f, 0.f, 0.f, 0.f, 0.f, 0.f, 0.f, 0.f};
  v8f accRe[2] = { zero, zero };
  v8f accIm[2] = { zero, zero };

  for (int k = 0; k < 32; k += 4) {
    int kb = k + 2 * half;             // A/B fragment: lanes 16-31 hold K=k+2,k+3
    int r0, r1;
    if (MODE_A) { r0 = kb * 32 + g; r1 = (kb + 1) * 32 + g; }
    else {
      r0 = g * 32 + kb; r1 = r0 + 1;
      if (GRAY) { r0 ^= (r0 >> 1); r1 ^= (r1 >> 1); }   // row i of P*M = row gray(i) of M
    }
    v2f bre = { inre[r0 * 1024 + col], inre[r1 * 1024 + col] };
    v2f bim = { 0.f, 0.f };
    if (!REAL_IN) bim = (v2f){ inim[r0 * 1024 + col], inim[r1 * 1024 + col] };

#pragma unroll
    for (int m = 0; m < 2; ++m) {
      int row = m * 16 + l15;
      v2f ar = { gsr[row][kb], gsr[row][kb + 1] };
      v2f ai = { gsi[row][kb], gsi[row][kb + 1] };
      accRe[m] = __builtin_amdgcn_wmma_f32_16x16x4_f32(false, ar, false, bre,
                                                       (short)0, accRe[m], false, false);
      accIm[m] = __builtin_amdgcn_wmma_f32_16x16x4_f32(false, ai, false, bre,
                                                       (short)0, accIm[m], false, false);
      if (!REAL_IN) {
        v2f ain = -ai;   // f32 WMMA has no A-neg modifier; negate fragment in VALU
        accRe[m] = __builtin_amdgcn_wmma_f32_16x16x4_f32(false, ain, false, bim,
                                                         (short)0, accRe[m], false, false);
        accIm[m] = __builtin_amdgcn_wmma_f32_16x16x4_f32(false, ar, false, bim,
                                                         (short)0, accIm[m], false, false);
      }
    }
  }

#pragma unroll
  for (int m = 0; m < 2; ++m)
#pragma unroll
    for (int v = 0; v < 8; ++v) {
      int rl = m * 16 + v + 8 * half;               // D layout: VGPR v <-> M=v (+8 upper half)
      int row = MODE_A ? (rl * 32 + g) : (g * 32 + rl);
      outre[row * 1024 + col] = accRe[m][v];
      outim[row * 1024 + col] = accIm[m][v];
    }
}

// ---------- kernel 4: result = sum_p z(gray^-1(p)) * <E'[p,:], W'[p,:]>  ----------
// z depends on bit4 of logical i = prefix-xor(p): bit4(i) = parity(p & 0x3F0)
__global__ __launch_bounds__(256) void reduce_rows(const float* __restrict__ Ere,
                                                   const float* __restrict__ Eim,
                                                   const float* __restrict__ Wre,
                                                   const float* __restrict__ Wim,
                                                   float* __restrict__ partials) {
  int p = blockIdx.x, tid = threadIdx.x;
  const float* er = Ere + (size_t)p * 1024;
  const float* ei = Eim + (size_t)p * 1024;
  const float* wr = Wre + (size_t)p * 1024;
  const float* wi = Wim + (size_t)p * 1024;
  float s = 0.f;
  for (int c = tid; c < 1024; c += 256)
    s += er[c] * wr[c] + ei[c] * wi[c];
  __shared__ float sm[256];
  sm[tid] = s;
  __syncthreads();
  for (int off = 128; off > 0; off >>= 1) {
    if (tid < off) sm[tid] += sm[tid + off];
    __syncthreads();
  }
  if (tid == 0) {
    float z = (__popc(p & 0x3F0) & 1) ? -1.f : 1.f;
    partials[p] = z * sm[0];
  }
}

__global__ __launch_bounds__(1024) void reduce_final(const float* __restrict__ partials,
                                                     float* __restrict__ out) {
  __shared__ float sm[1024];
  int t = threadIdx.x;
  sm[t] = partials[t];
  __syncthreads();
  for (int off = 512; off > 0; off >>= 1) {
    if (t < off) sm[t] += sm[t + off];
    __syncthreads();
  }
  if (t == 0) out[0] = sm[0];
}

// ---------- launch ----------
extern "C" void kernel_launch(void* const* d_in, const int* in_sizes, int n_in,
                              void* d_out, int out_size, void* d_ws, size_t ws_size,
                              hipStream_t stream) {
  const float* X = (const float*)d_in[0];      // 1024*1024 f32
  const float* W = (const float*)d_in[1];      // 30 f32

  float* ws = (float*)d_ws;
  float* gates    = ws;            // 8192 floats (A1,B1,A2,B2 re/im)
  float* partials = ws + 8192;     // 1024 floats
  float* planes   = ws + 16384;    // 6 planes x 1M floats
  auto P = [&](int i) { return planes + (size_t)i * (1u << 20); };
  float* A1re = gates + 0,    *A1im = gates + 1024;
  float* B1re = gates + 2048, *B1im = gates + 3072;
  float* A2re = gates + 4096, *A2im = gates + 5120;
  float* B2re = gates + 6144, *B2im = gates + 7168;

  build_gates<<<1, 256, 0, stream>>>(W, gates);
  // E' = T2 * P * T1  (outer P folded into the reduction)
  build_T1<<<4096, 256, 0, stream>>>(gates, P(0), P(1));                        // T1
  dim3 grid(32, 16), blk(128);
  apply_gate<0, 1, 0><<<grid, blk, 0, stream>>>(P(2), P(3), P(0), P(1), B2re, B2im); // (I(x)B2)*P*T1
  apply_gate<1, 0, 0><<<grid, blk, 0, stream>>>(P(4), P(5), P(2), P(3), A2re, A2im); // E'
  // W' = T2 * P * T1 * X
  apply_gate<0, 0, 1><<<grid, blk, 0, stream>>>(P(0), P(1), X, X, B1re, B1im);       // (I(x)B1)*X
  apply_gate<1, 0, 0><<<grid, blk, 0, stream>>>(P(2), P(3), P(0), P(1), A1re, A1im); // T1*X
  apply_gate<0, 1, 0><<<grid, blk, 0, stream>>>(P(0), P(1), P(2), P(3), B2re, B2im); // (I(x)B2)*P*T1*X
  apply_gate<1, 0, 0><<<grid, blk, 0, stream>>>(P(2), P(3), P(0), P(1), A2re, A2im); // W'
  // result = Re<E', Z_perm W'>
  reduce_rows<<<1024, 256, 0, stream>>>(P(4), P(5), P(2), P(3), partials);
  reduce_final<<<1, 1024, 0, stream>>>(partials, (float*)d_out);
}